// Loss_67010079752779
// MI455X (gfx1250) — compile-verified
//
#include <hip/hip_runtime.h>
#include <hip/hip_bf16.h>

typedef float v2f __attribute__((ext_vector_type(2)));
typedef float v8f __attribute__((ext_vector_type(8)));

#define KCH 80
#define HWPIX 197505          // 385*513
#define NIMG 8
#define TOTPIX (NIMG * HWPIX) // 1,580,040 pixels
#define EPSF 1e-11f
#define MAIN_BLOCKS 1544      // ~4 pixels/thread, 12k waves for latency hiding
#define BLOCK 256             // 8 wave32 per workgroup

// ---------------- ws init: per-image tmax slots ----------------
__global__ void init_ws(int* __restrict__ tmax) {
    if (threadIdx.x < NIMG) tmax[threadIdx.x] = -1;
}

// ---------------- per-image max(target) ----------------
__global__ void tmax_kernel(const int* __restrict__ tgt, int* __restrict__ tmax) {
    const int n = blockIdx.y;
    const int* img = tgt + (long long)n * HWPIX;
    int local = -1;
    for (int i = blockIdx.x * blockDim.x + threadIdx.x; i < HWPIX;
         i += gridDim.x * blockDim.x)
        local = max(local, img[i]);
    // wave32 reduce
    for (int off = 16; off > 0; off >>= 1)
        local = max(local, __shfl_down(local, off, 32));
    __shared__ int sred[BLOCK / 32];
    const int wave = threadIdx.x >> 5;
    if ((threadIdx.x & 31) == 0) sred[wave] = local;
    __syncthreads();
    if (threadIdx.x == 0) {
        int m = sred[0];
#pragma unroll
        for (int i = 1; i < BLOCK / 32; ++i) m = max(m, sred[i]);
        atomicMax(&tmax[n], m);
    }
}

// ---------------- main streaming kernel: one log per element ----------------
__global__ void ll_kernel(const float* __restrict__ mo, const int* __restrict__ tgt,
                          const int* __restrict__ tmax, float* __restrict__ partials) {
    const int tid = blockIdx.x * blockDim.x + threadIdx.x;
    const int stride = gridDim.x * blockDim.x;
    float acc = 0.0f;
    for (int p = tid; p < TOTPIX; p += stride) {
        const int n = p / HWPIX;
        const int phw = p - n * HWPIX;
        const int t = tgt[p];
        const int tm = tmax[n];
        const float* base = mo + (long long)n * KCH * HWPIX + phw;
        // k = 0 always has accum >= 1 -> sel = pos
        const float pos0 = __builtin_nontemporal_load(base);
        float ll = __logf(pos0 + EPSF);
#pragma unroll 4
        for (int k = 1; k < KCH; ++k) {
            const float pos = __builtin_nontemporal_load(base + (long long)k * HWPIX);
            const float sel = (k <= t) ? pos : (1.0f - pos);
            ll += __logf(sel + EPSF);
        }
        // accum[...,0]==2 iff t == tmax-1: add log(p0+e) - log(1-p0+e)
        if (t == tm - 1)
            ll += __logf(pos0 + EPSF) - __logf(1.0f - pos0 + EPSF);
        acc += ll;
    }
    // wave32 reduce then LDS cross-wave reduce
    for (int off = 16; off > 0; off >>= 1)
        acc += __shfl_down(acc, off, 32);
    __shared__ float sred[BLOCK / 32];
    const int wave = threadIdx.x >> 5;
    if ((threadIdx.x & 31) == 0) sred[wave] = acc;
    __syncthreads();
    if (threadIdx.x == 0) {
        float s = sred[0];
#pragma unroll
        for (int i = 1; i < BLOCK / 32; ++i) s += sred[i];
        partials[blockIdx.x] = s;
    }
}

// ---------------- final reduce: single wave32, WMMA f32 16x16x4 ----------------
// A VGPR0 = per-lane sum s, VGPR1 = 0  =>  A[m][0]=s(m), A[m][2]=s(m+16), K=1,3 zero.
// B = ones(4x16)  =>  D[m][n] = s(m) + s(m+16) for all n.
// D layout: VGPR j, lanes 0-15 hold D[j][lane]; lanes 16-31 hold D[j+8][lane-16].
// In-lane sum of c[0..7]: lane 0 -> sum of pairsum(0..7), lane 16 -> pairsum(8..15).
__global__ void final_kernel(const float* __restrict__ partials, int n,
                             float* __restrict__ out, float scale) {
    const int lane = threadIdx.x; // launched with exactly 32 threads, EXEC all-ones
    float s = 0.0f;
    for (int i = lane; i < n; i += 32) s += partials[i];
    v2f a; a.x = s;    a.y = 0.0f;
    v2f b; b.x = 1.0f; b.y = 1.0f;
    v8f c = {};
    c = __builtin_amdgcn_wmma_f32_16x16x4_f32(false, a, false, b, (short)0, c,
                                              false, false);
    const float r = c[0] + c[1] + c[2] + c[3] + c[4] + c[5] + c[6] + c[7];
    const float total = __shfl(r, 0, 32) + __shfl(r, 16, 32);
    if (lane == 0) out[0] = total * scale;
}

extern "C" void kernel_launch(void* const* d_in, const int* in_sizes, int n_in,
                              void* d_out, int out_size, void* d_ws, size_t ws_size,
                              hipStream_t stream) {
    const float* mo = (const float*)d_in[0];  // model_output [8,80,385,513] f32
    const int* tgt = (const int*)d_in[1];     // target [8,385,513] i32
    int* tmax = (int*)d_ws;                               // 8 ints
    float* partials = (float*)((char*)d_ws + 256);        // MAIN_BLOCKS floats
    float* out = (float*)d_out;

    init_ws<<<1, 32, 0, stream>>>(tmax);
    tmax_kernel<<<dim3(25, NIMG), BLOCK, 0, stream>>>(tgt, tmax);
    ll_kernel<<<MAIN_BLOCKS, BLOCK, 0, stream>>>(mo, tgt, tmax, partials);
    const float scale = -1.0f / ((float)KCH * (float)TOTPIX);
    final_kernel<<<1, 32, 0, stream>>>(partials, MAIN_BLOCKS, out, scale);
}